// RoINet_77249281786272
// MI455X (gfx1250) — compile-verified
//
#include <hip/hip_runtime.h>
#include <float.h>

// RoIPool(4,7,1/16) for feature [2,256,38,50] f32, rois [R,5] f32.
// Compute-trivial, L2-resident problem -> one-pass kernel, one block per
// (ROI, 32-channel slab). CDNA5 path: async global->LDS DMA staging of the
// ROI's bounding rectangle (ASYNCcnt), bins served from LDS.

#define B_     2
#define C_     256
#define H_     38
#define W_     50
#define PH_    4
#define PW_    7
#define SCALE_ (1.0f / 16.0f)

#define CH_PER_BLK 32
#define CGROUPS    (C_ / CH_PER_BLK)   // 8
#define THREADS    256
#define REG_H_CAP  20                  // expected worst regH ~13
#define REG_W_CAP  24                  // expected worst regW ~17
#define CH_STRIDE  (REG_H_CAP * REG_W_CAP + 1)   // 481 floats, de-conflicted
// LDS: 32 * 481 * 4 = 61,568 bytes (< 64KB; WGP has 320KB)

typedef __attribute__((address_space(1))) int gint_t;
typedef __attribute__((address_space(3))) int lint_t;

__global__ __launch_bounds__(THREADS)
void roi_pool_kernel(const float* __restrict__ feat,
                     const float* __restrict__ rois,
                     float* __restrict__ out,
                     int R)
{
    __shared__ float smem[CH_PER_BLK * CH_STRIDE];

    const int bid = blockIdx.x;
    const int r   = bid / CGROUPS;
    const int cg  = bid - r * CGROUPS;
    const int c0  = cg * CH_PER_BLK;
    if (r >= R) return;

    // ---- ROI params (uniform over the block; loads broadcast from L2) ----
    const float* roi = rois + (size_t)r * 5;
    const int b  = (int)roi[0];
    const int sw = (int)floorf(roi[1] * SCALE_ + 0.5f);   // round-half-up
    const int sh = (int)floorf(roi[2] * SCALE_ + 0.5f);
    const int ew = (int)floorf(roi[3] * SCALE_ + 0.5f);
    const int eh = (int)floorf(roi[4] * SCALE_ + 0.5f);
    const int roi_w = max(ew - sw + 1, 1);
    const int roi_h = max(eh - sh + 1, 1);
    const float bin_h = (float)roi_h / (float)PH_;
    const float bin_w = (float)roi_w / (float)PW_;

    // Union of all bins = one rectangle per channel (hend(PH-1)=clip(sh+roi_h))
    const int h0 = min(max(sh, 0), H_);
    const int h1 = min(max(sh + roi_h, 0), H_);
    const int w0 = min(max(sw, 0), W_);
    const int w1 = min(max(sw + roi_w, 0), W_);
    const int regH = h1 - h0;
    const int regW = w1 - w0;
    const int tileN = regH * regW;

    const bool useLds = (regH > 0) && (regW > 0) &&
                        (regH <= REG_H_CAP) && (regW <= REG_W_CAP);

    const size_t featBase = ((size_t)b * C_ + c0) * (size_t)(H_ * W_);

    // ---- Stage ROI rectangle x 32 channels into LDS via async DMA ----
    if (useLds) {
        for (int ch = 0; ch < CH_PER_BLK; ++ch) {
            const float* gsrc = feat + featBase + (size_t)ch * (H_ * W_);
            float* ldst = &smem[ch * CH_STRIDE];
            for (int rem = threadIdx.x; rem < tileN; rem += THREADS) {
                const int hh  = rem / regW;
                const int wwi = rem - hh * regW;
                const float* gp = gsrc + (size_t)(h0 + hh) * W_ + (w0 + wwi);
#if __has_builtin(__builtin_amdgcn_global_load_async_to_lds_b32)
                __builtin_amdgcn_global_load_async_to_lds_b32(
                    (gint_t*)(void*)gp,
                    (lint_t*)(void*)(ldst + rem),
                    0, 0);
#else
                ldst[rem] = *gp;
#endif
            }
        }
#if __has_builtin(__builtin_amdgcn_s_wait_asynccnt)
        __builtin_amdgcn_s_wait_asynccnt(0);
#endif
        __syncthreads();   // uniform branch: whole block takes same path
    }

    // ---- 32 channels x 28 bins = 896 outputs per block ----
    const int OUT_PER_BLK = CH_PER_BLK * PH_ * PW_;
    for (int idx = threadIdx.x; idx < OUT_PER_BLK; idx += THREADS) {
        const int ch  = idx / (PH_ * PW_);
        const int bin = idx - ch * (PH_ * PW_);
        const int ph  = bin / PW_;
        const int pw  = bin - ph * PW_;

        const int hs = min(max(sh + (int)floorf((float)ph       * bin_h), 0), H_);
        const int he = min(max(sh + (int)ceilf ((float)(ph + 1) * bin_h), 0), H_);
        const int ws = min(max(sw + (int)floorf((float)pw       * bin_w), 0), W_);
        const int we = min(max(sw + (int)ceilf ((float)(pw + 1) * bin_w), 0), W_);

        float v = 0.0f;                      // empty bin -> 0 (matches ref)
        if (he > hs && we > ws) {
            float m = -FLT_MAX;
            if (useLds) {
                const float* base = &smem[ch * CH_STRIDE];
                for (int h = hs; h < he; ++h) {
                    const float* row = base + (h - h0) * regW - w0;
                    for (int w = ws; w < we; ++w) m = fmaxf(m, row[w]);
                }
            } else {
                const float* fb = feat + featBase + (size_t)ch * (H_ * W_);
                for (int h = hs; h < he; ++h) {
                    const float* row = fb + (size_t)h * W_;
                    __builtin_prefetch(row + ws, 0, 1);  // global_prefetch_b8
                    for (int w = ws; w < we; ++w) m = fmaxf(m, row[w]);
                }
            }
            v = m;
        }
        out[(((size_t)r * C_ + (c0 + ch)) * PH_ + ph) * PW_ + pw] = v;
    }
}

extern "C" void kernel_launch(void* const* d_in, const int* in_sizes, int n_in,
                              void* d_out, int out_size, void* d_ws, size_t ws_size,
                              hipStream_t stream) {
    const float* feat = (const float*)d_in[0];   // [2,256,38,50] f32
    const float* rois = (const float*)d_in[1];   // [R,5] f32
    float* out = (float*)d_out;                  // [R,256,4,7] f32
    const int R = in_sizes[1] / 5;               // 256
    const int blocks = R * CGROUPS;              // 2048
    roi_pool_kernel<<<blocks, THREADS, 0, stream>>>(feat, rois, out, R);
}